// S42DLayer_18373870092726
// MI455X (gfx1250) — compile-verified
//
#include <hip/hip_runtime.h>
#include <math.h>

// ---------------------------------------------------------------------------
// S4-2D layer for gfx1250 (MI455X), wave32.
//   stage 1: generate kd0 (D,H) and kd1 (D,W) via Cauchy + inverse DFT (VALU)
//   stage 2: causal conv along axis1 as per-channel triangular-Toeplitz matmul
//            using V_WMMA_F32_16X16X4_F32, output written axis-swapped
//   stage 3: same kernel again along the other axis, fusing the D*u skip.
// Data movement: double-buffered 16-wide K stages through LDS via
// GLOBAL_LOAD_ASYNC_TO_LDS_B128 (ASYNCcnt), software-pipelined against the
// WMMA chain. Kernel rows are stored TIME-REVERSED with a zeroed tail so the
// Toeplitz A-fragment is a branch-free ascending ds_load_2addr pair (no
// conditional loads, no register swaps).
// ---------------------------------------------------------------------------

#define DM   256   // D_MODEL
#define NS   64    // N_STATE
#define SL   128   // SEQ_H == SEQ_W
#define BT   4     // BATCH

typedef float v2f __attribute__((ext_vector_type(2)));
typedef float v8f __attribute__((ext_vector_type(8)));

#if __has_builtin(__builtin_amdgcn_global_load_async_to_lds_b128)
#define USE_ASYNC_LDS 1
// The builtin's params are int4 pointers (GCC vector_size spelling) in the
// global (addrspace(1)) and shared (addrspace(3)) address spaces.
typedef int i4 __attribute__((vector_size(16)));
typedef __attribute__((address_space(1))) void* as1_void;
typedef __attribute__((address_space(3))) void* as3_void;
typedef __attribute__((address_space(1))) i4*   as1_i4;
typedef __attribute__((address_space(3))) i4*   as3_i4;
__device__ __forceinline__ as1_i4 to_global_i4(const void* p) {
  return (as1_i4)(as1_void)(p);
}
__device__ __forceinline__ as3_i4 to_lds_i4(void* p) {
  return (as3_i4)(as3_void)(p);
}
#endif

// ------------------------------ kernel generation ---------------------------
// grid = DM blocks, block = SL threads (one thread per frequency / time step)
__global__ __launch_bounds__(SL) void s4_gen_kernel(
    const float* __restrict__ p,   // (NS,2)
    const float* __restrict__ q,   // (NS,2)
    const float* __restrict__ lam, // (NS,2)
    const float* __restrict__ Bm,  // (DM,NS,2)
    const float* __restrict__ Ct,  // (DM,NS,2)
    const float* __restrict__ ls,  // (DM,)
    float* __restrict__ kd)        // (DM,SL) out
{
  __shared__ float s_re[SL];
  __shared__ float s_im[SL];
  const int d = blockIdx.x;
  const int l = threadIdx.x;
  const float PI = 3.14159265358979323846f;

  // omega_l = exp(-2*pi*i*l/SL)
  float ang = -2.f * PI * (float)l / (float)SL;
  float wr = __cosf(ang), wi = __sinf(ang);
  // c = 2/(1+omega)
  float opr = 1.f + wr, opi = wi;
  float opn = opr * opr + opi * opi;
  float cr = 2.f * opr / opn, ci = -2.f * opi / opn;
  // g = (2/step)*(1-omega)/(1+omega)
  float omr = 1.f - wr, omi = -wi;
  float fr = (omr * opr + omi * opi) / opn;
  float fi = (omi * opr - omr * opi) / opn;
  float scale = 2.f / __expf(ls[d]);
  float gr = scale * fr, gi = scale * fi;

  float k00r = 0, k00i = 0, k01r = 0, k01i = 0;
  float k10r = 0, k10i = 0, k11r = 0, k11i = 0;
  for (int n = 0; n < NS; ++n) {
    float lr = lam[2 * n], li = lam[2 * n + 1];
    float dr = gr - lr, di = gi - li;
    float dn = dr * dr + di * di;
    float idr = dr / dn, idi = -di / dn;                 // 1/denom
    float pr = p[2 * n], pim = p[2 * n + 1];
    float qr = q[2 * n], qi = -q[2 * n + 1];             // conj(q)
    float br = Bm[(d * NS + n) * 2], bi = Bm[(d * NS + n) * 2 + 1];
    float ctr = Ct[(d * NS + n) * 2], cti = -Ct[(d * NS + n) * 2 + 1]; // conj(Ct)
    float t0r = ctr * br - cti * bi, t0i = ctr * bi + cti * br;   // a0*B
    float t1r = ctr * pr - cti * pim, t1i = ctr * pim + cti * pr; // a0*p
    float t2r = qr * br - qi * bi,   t2i = qr * bi + qi * br;     // a1*B
    float t3r = qr * pr - qi * pim,  t3i = qr * pim + qi * pr;    // a1*p
    k00r += t0r * idr - t0i * idi;  k00i += t0r * idi + t0i * idr;
    k01r += t1r * idr - t1i * idi;  k01i += t1r * idi + t1i * idr;
    k10r += t2r * idr - t2i * idi;  k10i += t2r * idi + t2i * idr;
    k11r += t3r * idr - t3i * idi;  k11i += t3r * idi + t3i * idr;
  }
  // at = c * (k00 - k01*k10/(1+k11))
  float er = 1.f + k11r, ei = k11i;
  float en = er * er + ei * ei;
  float ier = er / en, iei = -ei / en;
  float mr = k01r * k10r - k01i * k10i, mi = k01r * k10i + k01i * k10r;
  float sr = mr * ier - mi * iei, si = mr * iei + mi * ier;
  float ur = k00r - sr, ui = k00i - si;
  s_re[l] = cr * ur - ci * ui;
  s_im[l] = cr * ui + ci * ur;
  __syncthreads();

  // time-reversed iFFT == forward DFT / L:
  float acc = 0.f;
  const float w0 = 2.f * PI / (float)SL;
  for (int ll = 0; ll < SL; ++ll) {
    int m = (l * ll) & (SL - 1);   // exact modular angle
    float th = w0 * (float)m;
    acc += s_re[ll] * __cosf(th) + s_im[ll] * __sinf(th);
  }
  kd[d * SL + l] = acc / (float)SL;
}

// ------------------------------- causal conv --------------------------------
// X: (BT, SL, SL, DM), convolve along axis1 (s1) with kd (DM,SL).
// Y[b, s2, s1, d] = sum_{j<=s1} kd[d,j] * X[b, s1-j, s2, d]   (axis-swapped out)
// If skip_u != null: Y += skip_d[d] * skip_u at the same (swapped) index.
// grid = (SL/16, DM/16, BT), block = 512 (16 waves; wave w owns channel d0+w).
#define KROW 148   // reversed row: [0..127]=kd[127..0], [128..143]=0, pad to 148
__global__ __launch_bounds__(512) void s4_conv_kernel(
    const float* __restrict__ X,
    const float* __restrict__ kd,
    float* __restrict__ Y,
    const float* __restrict__ skip_u,
    const float* __restrict__ skip_d)
{
  // minor dim padded to 20 words: 80B rows stay 16B-aligned for b128 and
  // stride-20 is conflict-free across 64 LDS banks for the 16 reading lanes.
  __shared__ __align__(16) float s_kr[16][KROW];       // reversed kernels
  __shared__ __align__(16) float s_b[2][16][16][20];   // K-stage dbl-buffer
  __shared__ __align__(16) float s_out[16][16][20];    // [m][n][d] staging

  const int tid  = threadIdx.x;
  const int lane = tid & 31;
  const int wd   = tid >> 5;              // wave id == channel within tile
  const int n0   = blockIdx.x * 16;
  const int d0   = blockIdx.y * 16;
  const int b    = blockIdx.z;

  // ---- kernel rows, time-reversed: s_kr[d][i] = kd[d][127-i]; tail zeros
  // ---- cover out-of-range taps (j down to -16). One b128 per thread,
  // ---- component-reversed in registers (124-j keeps 16B alignment).
  if (tid < 256) s_kr[tid >> 4][128 + (tid & 15)] = 0.f;
  {
    int i  = tid * 4;                     // multiple of 4
    int dl = i >> 7;
    int j  = i & (SL - 1);
    float4 s = *(const float4*)&kd[(d0 + dl) * SL + j];
    float4 r; r.x = s.w; r.y = s.z; r.z = s.y; r.w = s.x;
    *(float4*)&s_kr[dl][124 - j] = r;     // word 127-j gets kd[j]
  }

  v8f acc[8];
#pragma unroll
  for (int mt = 0; mt < 8; ++mt)
#pragma unroll
    for (int r = 0; r < 8; ++r) acc[mt][r] = 0.f;

  const int m_lane = lane & 15;
  const int n_lane = lane & 15;
  const int kb     = (lane >> 4) << 1;    // 0 or 2 (f32 16x16x4 A/B layout)
  const int baseX  = b * SL * SL * DM;

  // element->coords within a 16x16x16 stage: dl = e&15, nn = (e>>4)&15, kk = e>>8
  const int e1  = tid * 4;                // kk 0..7
  const int e2  = e1 + 2048;              // kk 8..15
  const int dl1 = e1 & 15, nn1 = (e1 >> 4) & 15, kk1 = e1 >> 8;
  const int dl2 = e2 & 15, nn2 = (e2 >> 4) & 15, kk2 = e2 >> 8;

  // stage loader: 16(k) x 16(n) x 16(d) floats, 2 x b128 per thread
#define LOAD_STAGE(KS, BUF)                                                   \
  do {                                                                        \
    const int _k0 = (KS) * 16;                                                \
    const int _g1 = baseX + ((_k0 + kk1) * SL + (n0 + nn1)) * DM + d0 + dl1;  \
    const int _g2 = baseX + ((_k0 + kk2) * SL + (n0 + nn2)) * DM + d0 + dl2;  \
    ASYNC_OR_SYNC(_g1, _g2, BUF);                                             \
  } while (0)

#ifdef USE_ASYNC_LDS
#define ASYNC_OR_SYNC(G1, G2, BUF)                                            \
    __builtin_amdgcn_global_load_async_to_lds_b128(                           \
        to_global_i4(&X[G1]), to_lds_i4(&s_b[BUF][kk1][nn1][dl1]), 0, 0);     \
    __builtin_amdgcn_global_load_async_to_lds_b128(                           \
        to_global_i4(&X[G2]), to_lds_i4(&s_b[BUF][kk2][nn2][dl2]), 0, 0)
#define WAIT_ASYNC() asm volatile("s_wait_asynccnt 0x0" ::: "memory")
#else
#define ASYNC_OR_SYNC(G1, G2, BUF)                                            \
    *(float4*)&s_b[BUF][kk1][nn1][dl1] = *(const float4*)&X[G1];              \
    *(float4*)&s_b[BUF][kk2][nn2][dl2] = *(const float4*)&X[G2]
#define WAIT_ASYNC() ((void)0)
#endif

  LOAD_STAGE(0, 0);                       // prologue: stage 0 in flight

  for (int ks = 0; ks < 8; ++ks) {        // 16-wide K stages, pipelined
    WAIT_ASYNC();                         // own wave's stage-ks loads done
    __syncthreads();                      // => all waves' stage-ks data in LDS
    if (ks < 7) LOAD_STAGE(ks + 1, (ks + 1) & 1);  // overlap with compute
    const int k0s = ks * 16;

#pragma unroll
    for (int kc4 = 0; kc4 < 4; ++kc4) {   // four K=4 WMMA sub-chunks
      const int kloc = kc4 * 4 + kb;
      v2f bfrag;
      bfrag[0] = s_b[ks & 1][kloc][n_lane][wd];
      bfrag[1] = s_b[ks & 1][kloc + 1][n_lane][wd];
#pragma unroll
      for (int mt = 0; mt < 8; ++mt) {
        if (mt >= ks) {                   // wave-uniform triangular guard
          // i0 = 127 - j, ascending pair (kd[j], kd[j-1]); always in [0,143]
          int i0 = 127 - (mt * 16 + m_lane) + (k0s + kloc);
          v2f afrag;
          afrag[0] = s_kr[wd][i0];
          afrag[1] = s_kr[wd][i0 + 1];
          acc[mt] = __builtin_amdgcn_wmma_f32_16x16x4_f32(
              false, afrag, false, bfrag, (short)0, acc[mt], false, false);
        }
      }
    }
  }

  // ---- write back, axis-swapped, via LDS transpose; b128 global stores ----
#pragma unroll
  for (int mt = 0; mt < 8; ++mt) {
    __syncthreads();
#pragma unroll
    for (int r = 0; r < 8; ++r) {
      int m = r + ((lane >> 4) << 3);
      s_out[m][n_lane][wd] = acc[mt][r];
    }
    __syncthreads();
#pragma unroll
    for (int h = 0; h < 2; ++h) {         // e = 4*tid (+2048): 2 x b128/thread
      const int e  = tid * 4 + h * 2048;
      const int dl = e & 15, nn = (e >> 4) & 15, mm = e >> 8;
      const int s1 = mt * 16 + mm;
      const int s2 = n0 + nn;
      const int dg = d0 + dl;
      const int oi = ((b * SL + s2) * SL + s1) * DM + dg;
      float4 v = *(float4*)&s_out[mm][nn][dl];
      if (skip_u) {
        float4 su = *(const float4*)&skip_u[oi];
        float4 sd = *(const float4*)&skip_d[dg];
        v.x += sd.x * su.x; v.y += sd.y * su.y;
        v.z += sd.z * su.z; v.w += sd.w * su.w;
      }
      *(float4*)&Y[oi] = v;
    }
  }
}

// --------------------------------- launcher ---------------------------------
extern "C" void kernel_launch(void* const* d_in, const int* in_sizes, int n_in,
                              void* d_out, int out_size, void* d_ws, size_t ws_size,
                              hipStream_t stream) {
  const float* u    = (const float*)d_in[0];
  const float* p0   = (const float*)d_in[1];
  const float* q0   = (const float*)d_in[2];
  const float* lam0 = (const float*)d_in[3];
  const float* B0   = (const float*)d_in[4];
  const float* Ct0  = (const float*)d_in[5];
  const float* ls0  = (const float*)d_in[6];
  const float* p1   = (const float*)d_in[7];
  const float* q1   = (const float*)d_in[8];
  const float* lam1 = (const float*)d_in[9];
  const float* B1   = (const float*)d_in[10];
  const float* Ct1  = (const float*)d_in[11];
  const float* ls1  = (const float*)d_in[12];
  const float* Dv   = (const float*)d_in[13];

  const size_t kdElems = (size_t)DM * SL;
  const size_t y1Elems = (size_t)BT * SL * SL * DM;
  if (ws_size < (2 * kdElems + y1Elems) * sizeof(float)) return;

  float* kd0 = (float*)d_ws;
  float* kd1 = kd0 + kdElems;
  float* y1  = kd1 + kdElems;

  s4_gen_kernel<<<DM, SL, 0, stream>>>(p0, q0, lam0, B0, Ct0, ls0, kd0);
  s4_gen_kernel<<<DM, SL, 0, stream>>>(p1, q1, lam1, B1, Ct1, ls1, kd1);

  dim3 grid(SL / 16, DM / 16, BT);
  // pass 1: conv along H, u(B,H,W,D) -> y1(B,W,H,D)
  s4_conv_kernel<<<grid, 512, 0, stream>>>(u, kd0, y1, nullptr, nullptr);
  // pass 2: conv along W, y1(B,W,H,D) -> out(B,H,W,D), fused D*u skip
  s4_conv_kernel<<<grid, 512, 0, stream>>>(y1, kd1, (float*)d_out, u, Dv);
}